// AttentionLayer_64347200029420
// MI455X (gfx1250) — compile-verified
//
#include <hip/hip_runtime.h>
#include <hip/hip_bf16.h>
#include <math.h>

typedef __attribute__((ext_vector_type(2))) float v2f;
typedef __attribute__((ext_vector_type(8))) float v8f;

static constexpr int kB  = 16;
static constexpr int kTE = 512;
static constexpr int kTD = 128;
static constexpr int kH  = 128;

// ---------------------------------------------------------------------------
// Kernel 1: Out[M,128] = A[M,128] x W[128,128]  (f32, V_WMMA_F32_16X16X4_F32)
// One wave (32 threads) computes a 16-row x 128-col strip: A fragment loaded
// once per K-step and reused across the 8 N-tiles.
// A frag (16x4, MxK): lanes 0-15 -> K=k0,k0+1 ; lanes 16-31 -> K=k0+2,k0+3.
// B frag (4x16, KxN): lane holds column N=lo, same half-wave K split.
// C/D: VGPR j holds row (j + 8*hi), col = lo.
// ---------------------------------------------------------------------------
__global__ __launch_bounds__(32)
void gemm_proj_kernel(const float* __restrict__ A, const float* __restrict__ W,
                      float* __restrict__ Out) {
    const int m0   = blockIdx.x * 16;
    const int lane = threadIdx.x & 31;
    const int hi   = lane >> 4;   // half-wave: selects K pair
    const int lo   = lane & 15;

    v8f acc[8];
#pragma unroll
    for (int nt = 0; nt < 8; ++nt) acc[nt] = (v8f)0.0f;

    const float* Arow = A + (size_t)(m0 + lo) * kH;
    for (int k0 = 0; k0 < kH; k0 += 4) {
        v2f a;
        a.x = Arow[k0 + 2 * hi];
        a.y = Arow[k0 + 2 * hi + 1];
#pragma unroll
        for (int nt = 0; nt < 8; ++nt) {
            v2f b;
            b.x = W[(size_t)(k0 + 2 * hi) * kH + nt * 16 + lo];
            b.y = W[(size_t)(k0 + 2 * hi + 1) * kH + nt * 16 + lo];
            acc[nt] = __builtin_amdgcn_wmma_f32_16x16x4_f32(
                false, a, false, b, (short)0, acc[nt], false, false);
        }
    }
#pragma unroll
    for (int nt = 0; nt < 8; ++nt) {
#pragma unroll
        for (int j = 0; j < 8; ++j) {
            Out[(size_t)(m0 + j + 8 * hi) * kH + nt * 16 + lo] = acc[nt][j];
        }
    }
}

// ---------------------------------------------------------------------------
// Kernel 2: fused scores + softmax for one (b,d) per block.
//   score[t] = sum_h V[h] * tanh(Wx[b,t,h] + Uh[b,d,h])   (never materialize
//   the [B,Td,Te,H] tensor). 8 waves; each wave reduces one t at a time with
//   a coalesced 128-float row read and a 5-step wave32 shuffle butterfly.
// Then block softmax over the 512 scores in LDS; normalized e -> d_out.
// ---------------------------------------------------------------------------
__global__ __launch_bounds__(256)
void score_softmax_kernel(const float* __restrict__ Wx, const float* __restrict__ Uh,
                          const float* __restrict__ V, float* __restrict__ Eout) {
    __shared__ float sU[kH];
    __shared__ float sV[kH];
    __shared__ float sScore[kTE];
    __shared__ float sRed[8];

    const int bd   = blockIdx.x;     // b*kTD + d
    const int b    = bd / kTD;
    const int tid  = threadIdx.x;
    const int lane = tid & 31;
    const int wave = tid >> 5;       // 0..7

    if (tid < kH) {
        sU[tid] = Uh[(size_t)bd * kH + tid];
        sV[tid] = V[tid];
    }
    __syncthreads();

    const float* WxB = Wx + (size_t)b * kTE * kH;
    for (int t = wave; t < kTE; t += 8) {
        const float* row = WxB + (size_t)t * kH;
        float s = 0.0f;
#pragma unroll
        for (int j = 0; j < 4; ++j) {
            int h = lane + 32 * j;
            s += sV[h] * tanhf(row[h] + sU[h]);   // v_tanh_f32 (TRANS pipe)
        }
#pragma unroll
        for (int off = 16; off > 0; off >>= 1)
            s += __shfl_xor(s, off, 32);          // wave32 butterfly
        if (lane == 0) sScore[t] = s;
    }
    __syncthreads();

    // ---- softmax over 512 scores ----
    float m = -INFINITY;
    for (int t = tid; t < kTE; t += 256) m = fmaxf(m, sScore[t]);
#pragma unroll
    for (int off = 16; off > 0; off >>= 1) m = fmaxf(m, __shfl_xor(m, off, 32));
    if (lane == 0) sRed[wave] = m;
    __syncthreads();
    m = sRed[0];
#pragma unroll
    for (int w = 1; w < 8; ++w) m = fmaxf(m, sRed[w]);

    float psum = 0.0f;
    for (int t = tid; t < kTE; t += 256) {
        float e = __expf(sScore[t] - m);
        sScore[t] = e;                 // same thread reads/writes same slots
        psum += e;
    }
#pragma unroll
    for (int off = 16; off > 0; off >>= 1) psum += __shfl_xor(psum, off, 32);
    __syncthreads();                   // everyone done reading sRed (max phase)
    if (lane == 0) sRed[wave] = psum;
    __syncthreads();
    float total = 0.0f;
#pragma unroll
    for (int w = 0; w < 8; ++w) total += sRed[w];
    const float inv = 1.0f / total;

    float* erow = Eout + (size_t)bd * kTE;
    for (int t = tid; t < kTE; t += 256) erow[t] = sScore[t] * inv;
}

// ---------------------------------------------------------------------------
// Kernel 3: c[b] = e[b] (128x512) x enc[b] (512x128), batched over b.
// One wave per 16x16 output tile, K=512 -> 128 WMMA f32 16x16x4 issues.
// ---------------------------------------------------------------------------
__global__ __launch_bounds__(32)
void context_gemm_kernel(const float* __restrict__ E, const float* __restrict__ Enc,
                         float* __restrict__ C) {
    const int b    = blockIdx.z;
    const int m0   = blockIdx.x * 16;  // d tile
    const int n0   = blockIdx.y * 16;  // h tile
    const int lane = threadIdx.x & 31;
    const int hi   = lane >> 4;
    const int lo   = lane & 15;

    const float* Eb   = E + (size_t)b * kTD * kTE;
    const float* EncB = Enc + (size_t)b * kTE * kH;
    const float* Arow = Eb + (size_t)(m0 + lo) * kTE;

    v8f acc = (v8f)0.0f;
    for (int k0 = 0; k0 < kTE; k0 += 4) {
        v2f a;
        a.x = Arow[k0 + 2 * hi];
        a.y = Arow[k0 + 2 * hi + 1];
        v2f bb;
        bb.x = EncB[(size_t)(k0 + 2 * hi) * kH + n0 + lo];
        bb.y = EncB[(size_t)(k0 + 2 * hi + 1) * kH + n0 + lo];
        acc = __builtin_amdgcn_wmma_f32_16x16x4_f32(
            false, a, false, bb, (short)0, acc, false, false);
    }

    float* Cb = C + (size_t)b * kTD * kH;
#pragma unroll
    for (int j = 0; j < 8; ++j)
        Cb[(size_t)(m0 + j + 8 * hi) * kH + n0 + lo] = acc[j];
}

// ---------------------------------------------------------------------------
extern "C" void kernel_launch(void* const* d_in, const int* in_sizes, int n_in,
                              void* d_out, int out_size, void* d_ws, size_t ws_size,
                              hipStream_t stream) {
    const float* enc = (const float*)d_in[0];  // [B,TE,H]
    const float* dec = (const float*)d_in[1];  // [B,TD,H]
    const float* Wa  = (const float*)d_in[2];  // [H,H]
    const float* Ua  = (const float*)d_in[3];  // [H,H]
    const float* Va  = (const float*)d_in[4];  // [H,1]

    float* Wx = (float*)d_ws;                      // B*TE*H floats (4 MB)
    float* Uh = Wx + (size_t)kB * kTE * kH;        // B*TD*H floats (1 MB)

    float* c_out = (float*)d_out;                  // [B,TD,H]  (first output)
    float* e_out = c_out + (size_t)kB * kTD * kH;  // [B,TD,TE] (second output)

    // Wx = enc @ W_a   (8192x128 rows)
    gemm_proj_kernel<<<dim3(kB * kTE / 16), dim3(32), 0, stream>>>(enc, Wa, Wx);
    // Uh = dec @ U_a   (2048x128 rows)
    gemm_proj_kernel<<<dim3(kB * kTD / 16), dim3(32), 0, stream>>>(dec, Ua, Uh);
    // fused tanh-score + softmax -> e_out
    score_softmax_kernel<<<dim3(kB * kTD), dim3(256), 0, stream>>>(Wx, Uh, Va, e_out);
    // c = e @ enc (batched WMMA)
    context_gemm_kernel<<<dim3(kTD / 16, kH / 16, kB), dim3(32), 0, stream>>>(e_out, enc, c_out);
}